// FeatureClassifierCAVE_70437463655137
// MI455X (gfx1250) — compile-verified
//
#include <hip/hip_runtime.h>

#define NUM_CLASSES 1000
#define NBINS 1024   // padded to a multiple of 16 so phase-2 tiles are uniform

typedef __attribute__((ext_vector_type(2))) float v2f;
typedef __attribute__((ext_vector_type(8))) float v8f;

// ---------------------------------------------------------------------------
// Phase 1: bandwidth-bound gather + LDS histogram.
// float4/int4 loads -> global_load_b128; shared-float atomicAdd -> ds_add_f32.
// Each block writes one zero-padded 1024-bin partial row to workspace.
// ---------------------------------------------------------------------------
__global__ void __launch_bounds__(256)
fc_hist_partial_kernel(const float* __restrict__ scores,
                       const int*   __restrict__ mgi,
                       const int*   __restrict__ mgc,
                       float*       __restrict__ partials,
                       int n)
{
    __shared__ float bins[NBINS];
    for (int c = threadIdx.x; c < NBINS; c += 256) bins[c] = 0.0f;
    __syncthreads();

    const int nvec = n >> 2;
    const float4* __restrict__ s4 = (const float4*)scores;
    const int4*   __restrict__ i4 = (const int4*)mgi;
    const int stride = gridDim.x * 256;

    for (int v = blockIdx.x * 256 + threadIdx.x; v < nvec; v += stride) {
        float4 s  = s4[v];
        int4   ix = i4[v];
        int c0 = mgc[ix.x];
        int c1 = mgc[ix.y];
        int c2 = mgc[ix.z];
        int c3 = mgc[ix.w];
        atomicAdd(&bins[c0], s.x);
        atomicAdd(&bins[c1], s.y);
        atomicAdd(&bins[c2], s.z);
        atomicAdd(&bins[c3], s.w);
    }

    // scalar tail (n not multiple of 4)
    const int tail = n & 3;
    if (blockIdx.x == 0 && threadIdx.x < tail) {
        int i = n - tail + threadIdx.x;
        atomicAdd(&bins[mgc[mgi[i]]], scores[i]);
    }
    __syncthreads();

    float* row = partials + (size_t)blockIdx.x * NBINS;
    for (int c = threadIdx.x; c < NBINS; c += 256) row[c] = bins[c];
}

// ---------------------------------------------------------------------------
// Phase 2: reduce G partial rows with V_WMMA_F32_16X16X4_F32.
// D = ones(16x4) * B(4x16) + C  ==> every D row = sum over 4 partial rows.
// One wave32 per 16-class column tile; 4 rows folded per WMMA issue.
// Because A is all-ones, any K-slot assignment inside B's VGPR layout sums
// identically; the lane->N mapping (N = lane & 15) matches the C/D layout.
// ---------------------------------------------------------------------------
__global__ void __launch_bounds__(32)
fc_reduce_wmma_kernel(const float* __restrict__ partials,
                      float*       __restrict__ out,
                      int g, float scale)
{
    const int tile   = blockIdx.x;           // 16-class column tile
    const int lane   = threadIdx.x;          // 0..31, EXEC all ones for WMMA
    const int col    = tile * 16 + (lane & 15);
    const int rowOff = lane >> 4;            // half-wave selects K row

    v2f a; a.x = 1.0f; a.y = 1.0f;           // all-ones 16x4 A matrix
    v8f c = {};

    const float* __restrict__ base = partials + col;
    for (int r = 0; r < g; r += 4) {
        v2f b;
        b.x = base[(size_t)(r + 0 + rowOff) * NBINS];   // rows r, r+1
        b.y = base[(size_t)(r + 2 + rowOff) * NBINS];   // rows r+2, r+3
        c = __builtin_amdgcn_wmma_f32_16x16x4_f32(
                /*neg_a=*/false, a, /*neg_b=*/false, b,
                /*c_mod=*/(short)0, c, /*reuse_a=*/false, /*reuse_b=*/false);
    }

    // D VGPR0, lanes 0-15: (M=0, N=lane) -> the column sums.
    if (lane < 16 && col < NUM_CLASSES)
        out[col] = c[0] * scale;
}

// ---------------------------------------------------------------------------
extern "C" void kernel_launch(void* const* d_in, const int* in_sizes, int n_in,
                              void* d_out, int out_size, void* d_ws, size_t ws_size,
                              hipStream_t stream)
{
    const float* scores = (const float*)d_in[0];
    const int*   mgi    = (const int*)d_in[1];
    const int*   mgc    = (const int*)d_in[2];
    float*       out    = (float*)d_out;
    const int n = in_sizes[0];
    const int m = in_sizes[2];
    const float scale = 1.0f / (float)m;

    // Partial-histogram rows live in workspace: G x 1024 floats.
    int g = 1024;                                        // 4 MB at default
    size_t need = (size_t)g * NBINS * sizeof(float);
    if (ws_size < need) {
        g = (int)(ws_size / (NBINS * sizeof(float)));
        g &= ~3;                                         // multiple of 4 for WMMA K
        if (g < 4) g = 4;
    }
    float* partials = (float*)d_ws;

    fc_hist_partial_kernel<<<g, 256, 0, stream>>>(scores, mgi, mgc, partials, n);

    const int tiles = (NUM_CLASSES + 15) / 16;           // 63 column tiles
    fc_reduce_wmma_kernel<<<tiles, 32, 0, stream>>>(partials, out, g, scale);
}